// SQTLayer_68289980006677
// MI455X (gfx1250) — compile-verified
//
#include <hip/hip_runtime.h>
#include <hip/hip_bf16.h>

// Problem constants (match reference)
#define BB 128
#define TT 512
#define DD 512
#define HH 200
#define HP 208    // H padded to multiple of 16
#define KK 9
#define MROWS (BB * TT)   // 65536

typedef __attribute__((ext_vector_type(2))) float v2f;
typedef __attribute__((ext_vector_type(8))) float v8f;

// ---------------------------------------------------------------------------
// Kernel 1: logits = relu(X @ W1 + b1) @ W2 + b2, fused, f32 WMMA 16x16x4.
// Grid: 512 blocks x 256 threads (8 waves). Each wave: one 16-row M block.
// LDS layouts are K-pair interleaved so every B fragment is one ds_load_b64.
// ---------------------------------------------------------------------------
__global__ __launch_bounds__(256) void fused_mlp_logits(
    const float* __restrict__ X,    // (65536, 512)
    const float* __restrict__ W1,   // (512, 200)
    const float* __restrict__ b1,   // (200)
    const float* __restrict__ W2,   // (200, 9)
    const float* __restrict__ b2,   // (9)
    float* __restrict__ logits)     // (65536, 9)
{
    // W1 stage (pair-interleaved) during GEMM1; hidden tiles (plain) for GEMM2.
    __shared__ float ldsA[128 * HP];        // 106,496 B
    // W2 stage, pair-interleaved, padded (208 x 16), lives whole kernel.
    __shared__ float ldsB[HP * 16];         //  13,312 B

    const int tid   = threadIdx.x;
    const int lane  = tid & 31;
    const int wave  = tid >> 5;
    const int mlo   = lane & 15;        // M index within tile (rows 0..15)
    const int kpair = lane >> 4;        // selects K pair {0,1} vs {2,3}
    const int rowbase = blockIdx.x * 128 + wave * 16;

    // ---- stage W2 once: element (h, n) -> ldsB[((h/2)*16 + n)*2 + (h&1)] ---
    for (int idx = tid; idx < HP * 16; idx += 256) {
        const int h = idx >> 4, n = idx & 15;
        const float v = (h < HH && n < KK) ? W2[h * KK + n] : 0.0f;
        ldsB[(((h >> 1) * 16 + n) << 1) | (h & 1)] = v;
    }

    v8f acc[13];
    #pragma unroll
    for (int nt = 0; nt < 13; ++nt) acc[nt] = (v8f)0.0f;

    // ---- GEMM1: loop K in chunks of 128 staged through LDS -----------------
    for (int kc = 0; kc < DD; kc += 128) {
        // stage W1[kc:kc+128, 0:208): element (k,c) -> ldsA[((k/2)*HP+c)*2+(k&1)]
        for (int idx = tid; idx < 128 * HP; idx += 256) {
            const int r = idx / HP, c = idx - r * HP;
            const float v = (c < HH) ? W1[(kc + r) * HH + c] : 0.0f;
            ldsA[(((r >> 1) * HP + c) << 1) | (r & 1)] = v;
        }
        __syncthreads();

        const float* xrow = X + (size_t)(rowbase + mlo) * DD + kc;
        for (int k0 = 0; k0 < 128; k0 += 4) {
            // A fragment: a[j] = X[m][kc + k0 + 2*kpair + j]  (streamed, NT)
            const float* ap = xrow + k0 + 2 * kpair;           // 8B aligned
            v2f a = __builtin_nontemporal_load((const v2f*)ap);
            // B fragments: one contiguous ds_load_b64 each
            const float* bp = ldsA + ((((k0 >> 1) + kpair) * HP) << 1);
            #pragma unroll
            for (int nt = 0; nt < 13; ++nt) {
                v2f bf = *(const v2f*)(bp + ((nt * 16 + mlo) << 1));
                acc[nt] = __builtin_amdgcn_wmma_f32_16x16x4_f32(
                    false, a, false, bf, (short)0, acc[nt], false, false);
            }
        }
        __syncthreads();
    }

    // ---- bias + relu, park hidden tile (16 x 208, plain layout) in LDS -----
    float* hb = ldsA + wave * (16 * HP);
    #pragma unroll
    for (int nt = 0; nt < 13; ++nt) {
        const int n = nt * 16 + mlo;
        const float bias = (n < HH) ? b1[n] : 0.0f;
        #pragma unroll
        for (int r = 0; r < 8; ++r) {
            const int m = r + 8 * kpair;     // C layout: lanes16-31 hold M=r+8
            float v = acc[nt][r] + bias;
            v = fmaxf(v, 0.0f);
            hb[m * HP + n] = v;
        }
    }
    // no barrier needed: each wave reads back only its own hidden region

    // ---- GEMM2: logits tile = hidden(16x208) @ W2pad(208x16) ---------------
    v8f lacc = (v8f)0.0f;
    const int n2 = mlo;                      // output class column (valid < 9)
    for (int h0 = 0; h0 < HP; h0 += 4) {
        const int hk = h0 + 2 * kpair;
        v2f a  = *(const v2f*)(hb + mlo * HP + hk);                 // contiguous
        v2f bf = *(const v2f*)(ldsB + ((((h0 >> 1) + kpair) * 16 + n2) << 1));
        lacc = __builtin_amdgcn_wmma_f32_16x16x4_f32(
            false, a, false, bf, (short)0, lacc, false, false);
    }

    if (n2 < KK) {
        const float bias2 = b2[n2];
        #pragma unroll
        for (int r = 0; r < 8; ++r) {
            const int m = r + 8 * kpair;
            logits[(size_t)(rowbase + m) * KK + n2] = lacc[r] + bias2;
        }
    }
}

// ---------------------------------------------------------------------------
// Kernel 2: CRF negative log-likelihood per batch. 128 blocks x 32 threads.
// ---------------------------------------------------------------------------
__global__ __launch_bounds__(32) void crf_nll(
    const float* __restrict__ logits,  // (B, T, K)
    const float* __restrict__ trans,   // (K, K)
    const int*   __restrict__ tags,    // (B, T)
    const int*   __restrict__ lengths, // (B)
    float* __restrict__ nll)           // (B)
{
    const int b    = blockIdx.x;
    const int lane = threadIdx.x;
    const int len  = lengths[b];
    const float* lg = logits + (size_t)b * TT * KK;
    const int*   tg = tags   + (size_t)b * TT;

    // unary score: sum_t<len logits[t, tags[t]]
    float part = 0.0f;
    for (int t = lane; t < TT; t += 32)
        if (t < len) part += lg[t * KK + tg[t]];
    // binary score: sum_{1<=t<len} trans[tags[t-1], tags[t]]
    for (int t = 1 + lane; t < TT; t += 32)
        if (t < len) part += trans[tg[t - 1] * KK + tg[t]];
    #pragma unroll
    for (int off = 16; off > 0; off >>= 1)
        part += __shfl_xor(part, off, 32);   // now every lane holds the sum

    // forward recursion: all lanes compute redundantly (free under SIMD)
    float tr[KK * KK];
    #pragma unroll
    for (int i = 0; i < KK * KK; ++i) tr[i] = trans[i];

    float alpha[KK];
    #pragma unroll
    for (int j = 0; j < KK; ++j) alpha[j] = lg[j];

    for (int t = 1; t < len; ++t) {
        const float* lt = lg + t * KK;
        float na[KK];
        #pragma unroll
        for (int j = 0; j < KK; ++j) {
            float m = alpha[0] + tr[0 * KK + j];
            #pragma unroll
            for (int i = 1; i < KK; ++i)
                m = fmaxf(m, alpha[i] + tr[i * KK + j]);
            float s = 0.0f;
            #pragma unroll
            for (int i = 0; i < KK; ++i)
                s += __expf(alpha[i] + tr[i * KK + j] - m);
            na[j] = m + __logf(s) + lt[j];
        }
        #pragma unroll
        for (int j = 0; j < KK; ++j) alpha[j] = na[j];
    }

    float m = alpha[0];
    #pragma unroll
    for (int j = 1; j < KK; ++j) m = fmaxf(m, alpha[j]);
    float s = 0.0f;
    #pragma unroll
    for (int j = 0; j < KK; ++j) s += __expf(alpha[j] - m);
    const float log_norm = m + __logf(s);

    if (lane == 0) nll[b] = -(part - log_norm);
}

// ---------------------------------------------------------------------------
// Kernel 3: deterministic mean over 128 NLLs -> scalar output.
// ---------------------------------------------------------------------------
__global__ __launch_bounds__(128) void mean_reduce(
    const float* __restrict__ nll, float* __restrict__ out)
{
    __shared__ float s[128];
    s[threadIdx.x] = nll[threadIdx.x];
    __syncthreads();
    #pragma unroll
    for (int off = 64; off > 0; off >>= 1) {
        if (threadIdx.x < off) s[threadIdx.x] += s[threadIdx.x + off];
        __syncthreads();
    }
    if (threadIdx.x == 0) out[0] = s[0] * (1.0f / (float)BB);
}

extern "C" void kernel_launch(void* const* d_in, const int* in_sizes, int n_in,
                              void* d_out, int out_size, void* d_ws, size_t ws_size,
                              hipStream_t stream) {
    const float* X    = (const float*)d_in[0];  // cont_repr (B,T,D)
    const float* W1   = (const float*)d_in[1];  // (D,H)
    const float* b1   = (const float*)d_in[2];  // (H)
    const float* W2   = (const float*)d_in[3];  // (H,K)
    const float* b2   = (const float*)d_in[4];  // (K)
    const float* trans= (const float*)d_in[5];  // (K,K)
    const int*   tags = (const int*)d_in[6];    // (B,T)
    const int*   lens = (const int*)d_in[7];    // (B)
    float* out = (float*)d_out;

    float* logits = (float*)d_ws;                       // 65536*9 floats
    float* nll    = logits + (size_t)MROWS * KK;        // 128 floats

    fused_mlp_logits<<<MROWS / 128, 256, 0, stream>>>(X, W1, b1, W2, b2, logits);
    crf_nll<<<BB, 32, 0, stream>>>(logits, trans, tags, lens, nll);
    mean_reduce<<<1, 128, 0, stream>>>(nll, out);
}